// AutomatedShiftFix_13546326851758
// MI455X (gfx1250) — compile-verified
//
#include <hip/hip_runtime.h>
#include <hip/hip_bf16.h>

// Problem constants from the reference.
#define BB 4
#define CC 3
#define HH 512
#define WW 512
#define NSH 9                    // shifts per axis: -4..4
#define PBLK 8                   // x-rows per workgroup
#define N_OUT (BB * CC * HH * WW)

// LDS strides chosen for bank-conflict-free access (64 banks x 4B):
//  - ref rows: 516 floats (516 mod 64 = 4 -> 16 consecutive rows hit 16
//    distinct banks when lanes read the same column of different rows)
//  - x rows: 532 floats = 16 left zeros + 512 data + 4 right zeros
//    (lanes read consecutive addresses -> conflict-free)
#define RSTRIDE 516
#define XSTRIDE 532
#define XPAD 16

typedef __attribute__((ext_vector_type(2))) float v2f;
typedef __attribute__((ext_vector_type(8))) float v8f;

// ---------------------------------------------------------------------------
// Kernel 0: zero the sims accumulator in workspace (4 batches x 256 floats).
// ---------------------------------------------------------------------------
__global__ void sims_init_kernel(float* __restrict__ g_sims) {
    int t = threadIdx.x;
    for (int e = t; e < BB * 256; e += 256) g_sims[e] = 0.0f;
}

// ---------------------------------------------------------------------------
// Kernel 1: WMMA correlation.
//   grid = (H/PBLK, C, B), block = 128 (4 waves of 32).
//   Wave-level: D[sy_idx, sx_idx] += A[sy_idx, j] * B[j, sx_idx] over K=512,
//   with A = x row p (shifted by sy, zero-padded), B = x_ref rows p+sx.
//
//   Only D[0..8][0..8] is ever read, and D[M,N] depends only on A row M and
//   B column N -- so A rows 9..15 / B cols 9..15 may hold garbage (finite
//   floats). This lets every lane load unconditionally (no EXEC masking):
//   A indices are kept in-bounds by the 16-wide zero pad, B row indices by a
//   branchless clamp.
//
// VGPR layouts per CDNA5 ISA 7.12.2:
//   A (16x4 f32): lanes 0-15 -> M=lane, {v0=K0, v1=K1}; lanes 16-31 -> {K2,K3}
//   B (4x16 f32): lanes hold N=lane%16; {v0,v1} = rows {K0,K1} (lanes<16)
//                 or {K2,K3} (lanes>=16)  [assumed symmetric with A]
//   C/D (16x16 f32): vgpr r -> M = r + 8*(lane>=16), N = lane%16
// ---------------------------------------------------------------------------
__global__ void corr_wmma_kernel(const float* __restrict__ xref,
                                 const float* __restrict__ x,
                                 float* __restrict__ g_sims) {
    __shared__ float s_ref[(PBLK + 8) * RSTRIDE];   // rows p0-4 .. p0+11
    __shared__ float s_x[PBLK * XSTRIDE];           // zero-padded x rows
    __shared__ float s_sims[256];

    const int p0 = blockIdx.x * PBLK;
    const int c  = blockIdx.y;
    const int b  = blockIdx.z;
    const int tid = threadIdx.x;
    const long base_bc = ((long)b * CC + c) * (long)(HH * WW);

    // Stage ref rows [p0-4, p0+12), zero-filled outside the image; pad cols.
    for (int idx = tid; idx < (PBLK + 8) * RSTRIDE; idx += 128) {
        int row = idx / RSTRIDE;
        int col = idx - row * RSTRIDE;
        int grow = p0 - 4 + row;
        float v = 0.0f;
        if (col < WW && grow >= 0 && grow < HH)
            v = xref[base_bc + (long)grow * WW + col];
        s_ref[idx] = v;
    }
    // Stage x rows [p0, p0+8): 16 zeros | 512 data | 4 zeros.
    for (int idx = tid; idx < PBLK * XSTRIDE; idx += 128) {
        int row = idx / XSTRIDE;
        int col = idx - row * XSTRIDE;
        int j = col - XPAD;
        float v = 0.0f;
        if (j >= 0 && j < WW) v = x[base_bc + (long)(p0 + row) * WW + j];
        s_x[idx] = v;
    }
    for (int e = tid; e < 256; e += 128) s_sims[e] = 0.0f;
    __syncthreads();

    const int wave = tid >> 5;
    const int lane = tid & 31;
    const int mn   = lane & 15;      // M (= sy index) for A, N (= sx index) for B
    const int half = lane >> 4;      // selects K pair {0,1} vs {2,3}

    v8f acc = {};                    // f32 accumulator carries over both rows
    for (int r = 0; r < 2; ++r) {
        const int p_local = wave * 2 + r;
        // A element K=kk lives at x col (kk - mn) -> LDS idx +XPAD.
        // Min index over mn<=15 is (0 - 15 + 16) = 1: always in bounds.
        const int abase = p_local * XSTRIDE + 2 * half - mn + XPAD;
        // B col mn uses ref row p_local + mn; clamp keeps mn>8 lanes in-bounds.
        const int rl = (p_local + mn) < (PBLK + 8 - 1) ? (p_local + mn)
                                                       : (PBLK + 8 - 1);
        const int bbase = rl * RSTRIDE + 2 * half;

        for (int k = 0; k < WW; k += 16) {
            v2f a[4], bv[4];
#pragma unroll
            for (int u = 0; u < 4; ++u) {
                int ji = abase + k + 4 * u;
                a[u].x = s_x[ji];
                a[u].y = s_x[ji + 1];
                int ri = bbase + k + 4 * u;
                bv[u].x = s_ref[ri];
                bv[u].y = s_ref[ri + 1];
            }
#pragma unroll
            for (int u = 0; u < 4; ++u)
                acc = __builtin_amdgcn_wmma_f32_16x16x4_f32(
                    false, a[u], false, bv[u], (short)0, acc, false, false);
        }
    }

    // Per-workgroup reduction of the 16x16 D tiles into s_sims.
    for (int r = 0; r < 8; ++r) {
        int e = (r + 8 * half) * 16 + mn;   // M*16 + N
        atomicAdd(&s_sims[e], acc[r]);
    }
    __syncthreads();
    for (int e = tid; e < 256; e += 128)
        atomicAdd(&g_sims[b * 256 + e], s_sims[e]);
}

// ---------------------------------------------------------------------------
// Kernel 2: argmax over the 9x9 grid in reference flat order (sx outer,
// sy inner, first max wins), store int shifts to ws and floats to d_out tail.
// ---------------------------------------------------------------------------
__global__ void argmax_kernel(const float* __restrict__ g_sims,
                              int* __restrict__ g_shifts,
                              float* __restrict__ out_tail) {
    int b = threadIdx.x;
    if (b >= BB) return;
    float best = -3.4e38f;
    int bsx = 0, bsy = 0;
    for (int sx = 0; sx < NSH; ++sx) {
        for (int sy = 0; sy < NSH; ++sy) {
            float v = g_sims[b * 256 + sy * 16 + sx];   // M=sy, N=sx
            if (v > best) { best = v; bsx = sx - 4; bsy = sy - 4; }
        }
    }
    g_shifts[b * 2 + 0] = bsx;
    g_shifts[b * 2 + 1] = bsy;
    out_tail[b * 2 + 0] = (float)bsx;
    out_tail[b * 2 + 1] = (float)bsy;
}

// ---------------------------------------------------------------------------
// Kernel 3: apply the winning shift (zero-padded roll).
// ---------------------------------------------------------------------------
__global__ void apply_shift_kernel(const float* __restrict__ x,
                                   const int* __restrict__ g_shifts,
                                   float* __restrict__ out) {
    long t = (long)blockIdx.x * 256 + threadIdx.x;
    if (t >= (long)N_OUT) return;
    int j  = (int)(t & (WW - 1));
    int i  = (int)((t >> 9) & (HH - 1));
    int cb = (int)(t >> 18);            // b*C + c
    int b  = cb / CC;
    int sx = g_shifts[b * 2 + 0];
    int sy = g_shifts[b * 2 + 1];
    int si = i - sx, sj = j - sy;
    float v = 0.0f;
    if (si >= 0 && si < HH && sj >= 0 && sj < WW)
        v = x[((long)cb << 18) + (long)si * WW + sj];
    out[t] = v;
}

extern "C" void kernel_launch(void* const* d_in, const int* in_sizes, int n_in,
                              void* d_out, int out_size, void* d_ws, size_t ws_size,
                              hipStream_t stream) {
    (void)in_sizes; (void)n_in; (void)out_size; (void)ws_size;
    const float* xref = (const float*)d_in[0];   // x_ref
    const float* x    = (const float*)d_in[1];   // x
    float* out        = (float*)d_out;

    float* g_sims  = (float*)d_ws;                       // 4*256 floats
    int*   g_shift = (int*)((char*)d_ws + 4096);         // 4*2 ints

    sims_init_kernel<<<1, 256, 0, stream>>>(g_sims);

    dim3 grid(HH / PBLK, CC, BB);                        // (64, 3, 4)
    corr_wmma_kernel<<<grid, 128, 0, stream>>>(xref, x, g_sims);

    argmax_kernel<<<1, 32, 0, stream>>>(g_sims, g_shift, out + (long)N_OUT);

    int nblk = (N_OUT + 255) / 256;
    apply_shift_kernel<<<nblk, 256, 0, stream>>>(x, g_shift, out);
}